// NCC_22874995818977
// MI455X (gfx1250) — compile-verified
//
#include <hip/hip_runtime.h>

// ---------------------------------------------------------------------------
// Local NCC (window 9^3) for [1,2,192,224,192] f32 volumes on gfx1250.
// Fused single-pass: separable box filter where the x/y 9-tap filters are
// expressed as f32 WMMA matmuls (V_WMMA_F32_16X16X4_F32) against a constant
// 0/1 banded matrix, and the z filter is a sliding running sum backed by a
// 9-slot LDS ring buffer. Deterministic two-stage reduction to the scalar.
// ---------------------------------------------------------------------------

typedef float v8f __attribute__((ext_vector_type(8)));
typedef float v2f __attribute__((ext_vector_type(2)));

#define Wd 192
#define Hd 224
#define Dd 192
#define Cd 2
#define TX 32          // x outputs per block
#define TY 16          // y outputs per block
#define CZ 48          // z outputs per block
#define NX (Wd / TX)   // 6
#define NY (Hd / TY)   // 14
#define NZ (Dd / CZ)   // 4
#define NBLK (NX * NY * NZ * Cd)   // 672
#define WSZ 729.0f
#define NTOT 16515072.0            // 1*2*192*224*192

// D = A(16x4,f32) * B(4x16,f32) + C(16x16,f32), wave32, all lanes active.
__device__ __forceinline__ v8f wmma4(v2f a, v2f b, v8f c) {
  return __builtin_amdgcn_wmma_f32_16x16x4_f32(
      /*neg_a=*/false, a, /*neg_b=*/false, b,
      /*c_mod=*/(short)0, c, /*reuse_a=*/false, /*reuse_b=*/false);
}

// One 16x16 output block of the 9-tap y-filter: D[x, y_out] = sum_k XF[k][x]*band[k,y_out]
__device__ __forceinline__ v8f yfilter_block(int t, int n, int kh,
                                             const float (*XF)[24][33]) {
  const int mb = t & 1, q = t >> 1;
  const int m = mb * 16 + n;
  v8f acc = {};
#pragma unroll
  for (int c6 = 0; c6 < 6; ++c6) {
    const int k0 = 4 * c6 + 2 * kh;
    v2f a;
    a.x = XF[q][k0][m];
    a.y = XF[q][k0 + 1][m];
    const int d0 = k0 - n;
    v2f b;
    b.x = (d0 >= 0 && d0 <= 8) ? 1.0f : 0.0f;
    b.y = (d0 >= -1 && d0 <= 7) ? 1.0f : 0.0f;
    acc = wmma4(a, b, acc);
  }
  return acc;
}

__device__ __forceinline__ void store_ring(v8f acc, int t, int n, int kh,
                                           int slot, float (*ZR)[9][16][33]) {
  const int mb = t & 1, q = t >> 1;
#pragma unroll
  for (int r = 0; r < 8; ++r) {
    const int x = mb * 16 + r + 8 * kh;
    ZR[q][slot][n][x] = acc[r];
  }
}

__global__ __launch_bounds__(128) void ncc_main(const float* __restrict__ gI,
                                                const float* __restrict__ gJ,
                                                float* __restrict__ partial) {
  __shared__ float sP[2][32][42];        // padded input slice (rows 24..31 stay 0)
  __shared__ float sXF[5][24][33];       // x-filtered slice, 5 quantities
  __shared__ float sZR[5][9][16][33];    // 9-slot z ring buffer
  __shared__ float sRed[128];

  const int tid = threadIdx.x;
  const int lane = tid & 31;
  const int wave = tid >> 5;
  const int n = lane & 15;   // WMMA column index (N)
  const int kh = lane >> 4;  // WMMA K-half selector

  const int bx = blockIdx.x;
  const int tx0 = (bx % NX) * TX;
  const int ty0 = (bx / NX) * TY;
  const int zc0 = blockIdx.y * CZ;
  const int ch = blockIdx.z;

  // Per-thread voxel ownership for the z running sums: 4 voxels each.
  const int ox = tid & 31;
  const int oy0 = (tid >> 5) * 4;

  float S[5][4];
#pragma unroll
  for (int q = 0; q < 5; ++q)
#pragma unroll
    for (int i = 0; i < 4; ++i) S[q][i] = 0.0f;

  float ccsum = 0.0f;

  // Zero pad rows 24..31 of sP once (x-filter M-block 1 reads them).
  for (int idx = tid; idx < 2 * 8 * 42; idx += 128) {
    const int a = idx / (8 * 42);
    const int rr = (idx / 42) % 8;
    const int cc = idx % 42;
    sP[a][24 + rr][cc] = 0.0f;
  }

  const size_t chBase = (size_t)ch * Dd * Hd * Wd;

  for (int it = 0; it < CZ + 8; ++it) {
    const int z = zc0 - 4 + it;
    const bool zv = (z >= 0) && (z < Dd);
    const int slot = it % 9;

    if (zv) {
      const size_t zBase = chBase + (size_t)z * (Hd * Wd);
      for (int idx = tid; idx < 24 * 40; idx += 128) {
        const int r = idx / 40;
        const int c = idx - r * 40;
        const int ya = ty0 + r - 4;
        const int xa = tx0 + c - 4;
        float vi = 0.0f, vj = 0.0f;
        if (ya >= 0 && ya < Hd && xa >= 0 && xa < Wd) {
          const size_t off = zBase + (size_t)ya * Wd + xa;
          vi = gI[off];
          vj = gJ[off];
        }
        sP[0][r][c] = vi;
        sP[1][r][c] = vj;
      }
    }
    __syncthreads();  // (B) padded slice ready

    if (zv) {
      // x-filter: each wave owns one (mb, nb) 16x16 output block, all 5 q.
      const int mb = wave & 1, nb = wave >> 1;
      const int m = mb * 16 + n;
      v8f acc0 = {}, acc1 = {}, acc2 = {}, acc3 = {}, acc4 = {};
#pragma unroll
      for (int c6 = 0; c6 < 6; ++c6) {
        const int k0 = 4 * c6 + 2 * kh;
        const int kx = nb * 16 + k0;
        const float pI0 = sP[0][m][kx], pI1 = sP[0][m][kx + 1];
        const float pJ0 = sP[1][m][kx], pJ1 = sP[1][m][kx + 1];
        const int d0 = k0 - n;
        v2f b;
        b.x = (d0 >= 0 && d0 <= 8) ? 1.0f : 0.0f;
        b.y = (d0 >= -1 && d0 <= 7) ? 1.0f : 0.0f;
        v2f a;
        a.x = pI0;        a.y = pI1;        acc0 = wmma4(a, b, acc0);
        a.x = pJ0;        a.y = pJ1;        acc1 = wmma4(a, b, acc1);
        a.x = pI0 * pI0;  a.y = pI1 * pI1;  acc2 = wmma4(a, b, acc2);
        a.x = pJ0 * pJ0;  a.y = pJ1 * pJ1;  acc3 = wmma4(a, b, acc3);
        a.x = pI0 * pJ0;  a.y = pI1 * pJ1;  acc4 = wmma4(a, b, acc4);
      }
#pragma unroll
      for (int r = 0; r < 8; ++r) {
        const int row = mb * 16 + r + 8 * kh;
        if (row < 24) {
          const int col = nb * 16 + n;
          sXF[0][row][col] = acc0[r];
          sXF[1][row][col] = acc1[r];
          sXF[2][row][col] = acc2[r];
          sXF[3][row][col] = acc3[r];
          sXF[4][row][col] = acc4[r];
        }
      }
    }
    __syncthreads();  // (C) XF ready; ring slot still holds slice z-9

    // Subtract the slice leaving the 9-window (slot written 9 iterations ago).
    if (it >= 9) {
#pragma unroll
      for (int i = 0; i < 4; ++i) {
        const int y = oy0 + i;
#pragma unroll
        for (int q = 0; q < 5; ++q) S[q][i] -= sZR[q][slot][y][ox];
      }
    }

    // y-filter: 10 (mb, q) block tasks over 4 waves (computed into registers).
    v8f ya0 = {}, ya1 = {}, ya2 = {};
    const bool has2 = (wave < 2);
    if (zv) {
      ya0 = yfilter_block(wave, n, kh, sXF);
      ya1 = yfilter_block(wave + 4, n, kh, sXF);
      if (has2) ya2 = yfilter_block(wave + 8, n, kh, sXF);
    }
    __syncthreads();  // (D) old-ring reads complete

    if (zv) {
      store_ring(ya0, wave, n, kh, slot, sZR);
      store_ring(ya1, wave + 4, n, kh, slot, sZR);
      if (has2) store_ring(ya2, wave + 8, n, kh, slot, sZR);
    } else {
#pragma unroll
      for (int i = 0; i < 4; ++i) {
        const int y = oy0 + i;
#pragma unroll
        for (int q = 0; q < 5; ++q) sZR[q][slot][y][ox] = 0.0f;
      }
    }
    __syncthreads();  // (E) new slice in ring

    if (zv) {
#pragma unroll
      for (int i = 0; i < 4; ++i) {
        const int y = oy0 + i;
#pragma unroll
        for (int q = 0; q < 5; ++q) S[q][i] += sZR[q][slot][y][ox];
      }
    }

    // Output slice zo = z - 4 is complete now.
    const int zo = z - 4;
    if (zo >= zc0 && zo < zc0 + CZ) {
      const float inv = 1.0f / WSZ;
#pragma unroll
      for (int i = 0; i < 4; ++i) {
        const float Is = S[0][i], Js = S[1][i];
        const float I2 = S[2][i], J2 = S[3][i], IJ = S[4][i];
        const float uI = Is * inv, uJ = Js * inv;
        const float cross = IJ - uJ * Is - uI * Js + uI * uJ * WSZ;
        const float Ivar = I2 - 2.0f * uI * Is + uI * uI * WSZ;
        const float Jvar = J2 - 2.0f * uJ * Js + uJ * uJ * WSZ;
        ccsum += (cross * cross) / (Ivar * Jvar + 1e-5f);
      }
    }
    __syncthreads();  // end-of-iteration: protects sP/sXF/ring for next slice
  }

  // Deterministic block reduction.
  sRed[tid] = ccsum;
  __syncthreads();
#pragma unroll
  for (int s = 64; s > 0; s >>= 1) {
    if (tid < s) sRed[tid] += sRed[tid + s];
    __syncthreads();
  }
  if (tid == 0) {
    const int blin =
        blockIdx.x + gridDim.x * (blockIdx.y + gridDim.y * blockIdx.z);
    partial[blin] = sRed[0];
  }
}

__global__ __launch_bounds__(256) void ncc_final(const float* __restrict__ partial,
                                                 float* __restrict__ out) {
  __shared__ double sr[256];
  double s = 0.0;
  for (int i = threadIdx.x; i < NBLK; i += 256) s += (double)partial[i];
  sr[threadIdx.x] = s;
  __syncthreads();
#pragma unroll
  for (int k = 128; k > 0; k >>= 1) {
    if ((int)threadIdx.x < k) sr[threadIdx.x] += sr[threadIdx.x + k];
    __syncthreads();
  }
  if (threadIdx.x == 0) out[0] = (float)(-sr[0] / NTOT);
}

extern "C" void kernel_launch(void* const* d_in, const int* in_sizes, int n_in,
                              void* d_out, int out_size, void* d_ws,
                              size_t ws_size, hipStream_t stream) {
  (void)in_sizes; (void)n_in; (void)out_size; (void)ws_size;
  const float* gI = (const float*)d_in[0];  // y_pred
  const float* gJ = (const float*)d_in[1];  // y_true
  float* partial = (float*)d_ws;

  dim3 grid(NX * NY, NZ, Cd);  // (84, 4, 2) = 672 blocks
  ncc_main<<<grid, 128, 0, stream>>>(gI, gJ, partial);
  ncc_final<<<1, 256, 0, stream>>>(partial, (float*)d_out);
}